// CostomAdaptiveAvgPool2D_41162966565201
// MI455X (gfx1250) — compile-verified
//
#include <hip/hip_runtime.h>

typedef __attribute__((ext_vector_type(2))) float v2f;
typedef __attribute__((ext_vector_type(8))) float v8f;

#define H_IN   224
#define W_IN   224
#define PLANE  (H_IN * W_IN)          // 50176 floats per (b,c) plane
#define NPLANE (16 * 256)             // 4096 planes
#define OUTSZ  7

// One block per (b,c) plane. 224 threads = 7 wave32s; wave o owns output row o
// (input rows 32o .. 32o+31). Fully-coalesced b128 streaming loads, per-lane
// float4 accumulation, then ONE v_wmma_f32_16x16x4_f32 per wave performs the
// entire 7-window cross-lane pooled reduction.
__global__ __launch_bounds__(224) void adaptive_pool7_kernel(
    const float* __restrict__ x, float* __restrict__ out) {

  const int plane = blockIdx.x;            // b*256 + c
  const int lane  = threadIdx.x & 31;
  const int o     = threadIdx.x >> 5;      // 0..6  (output row / wave id)

  const float4* base = (const float4*)(x + (size_t)plane * PLANE);
  const int rowChunks = W_IN / 4;          // 56 float4 per row

  float4 accA = make_float4(0.f, 0.f, 0.f, 0.f);  // window  lane>>3      (0..3)
  float4 accB = make_float4(0.f, 0.f, 0.f, 0.f);  // window 4 + (lane>>3) (4..6), lanes 0..23

  const int h0 = o * 32;
  #pragma unroll 8
  for (int r = 0; r < 32; ++r) {
    const float4* row = base + (size_t)(h0 + r) * rowChunks;
    // chunks 0..31: lanes read 512 contiguous bytes -> global_load_b128, coalesced
    float4 va = row[lane];
    accA.x += va.x; accA.y += va.y; accA.z += va.z; accA.w += va.w;
    // chunks 32..55: lanes 0..23 read the remaining 384 contiguous bytes
    if (lane < 24) {
      float4 vb = row[32 + lane];
      accB.x += vb.x; accB.y += vb.y; accB.z += vb.z; accB.w += vb.w;
    }
  }

  // Horizontal sums: each lane now holds one partial per window it touched.
  const float sa = (accA.x + accA.y) + (accA.z + accA.w);
  const float sb = (accB.x + accB.y) + (accB.z + accB.w);  // 0 for lanes >= 24

  // A-matrix (16x4 f32, ISA layout: VGPR0 = K0|K2, VGPR1 = K1|K3 split at lane 16):
  //   A[m,0]=sa(lanes 0-15)  A[m,2]=sa(lanes 16-31)
  //   A[m,1]=sb(lanes 0-15)  A[m,3]=sb(lanes 16-31)
  v2f A; A.x = sa; A.y = sb;

  // One-hot B (4x16): B[0,0]=1, B[2,1]=1, B[1,2]=1, B[3,3]=1 routes each K-slice
  // to its own D column so D[m,n] isolates one lane-group's data.
  const int n    = lane & 15;
  const int half = lane >> 4;
  v2f B;
  B.x = (half ? (n == 1) : (n == 0)) ? 1.0f : 0.0f;   // B[0,n] / B[2,n]
  B.y = (half ? (n == 3) : (n == 2)) ? 1.0f : 0.0f;   // B[1,n] / B[3,n]

  v8f C = {};
  // D = A x B : emits v_wmma_f32_16x16x4_f32 (EXEC all ones here)
  v8f D = __builtin_amdgcn_wmma_f32_16x16x4_f32(
      /*neg_a=*/false, A, /*neg_b=*/false, B,
      /*c_mod=*/(short)0, C, /*reuse_a=*/false, /*reuse_b=*/false);

  // Sum the 8 C/D VGPRs lane-locally: lanes 0-15 get sum over M=0..7,
  // lanes 16-31 get sum over M=8..15 -> exactly the 8-lane group sums.
  const float s = ((D[0] + D[1]) + (D[2] + D[3])) + ((D[4] + D[5]) + (D[6] + D[7]));

  // Window p lives at lane {0,16,1,17,2,18,3}: p = 2*n + half, valid for n<4, p<7.
  const int p = 2 * n + half;
  if (n < 4 && p < OUTSZ) {
    out[(size_t)plane * (OUTSZ * OUTSZ) + o * OUTSZ + p] = s * (1.0f / 1024.0f);
  }
}

extern "C" void kernel_launch(void* const* d_in, const int* in_sizes, int n_in,
                              void* d_out, int out_size, void* d_ws, size_t ws_size,
                              hipStream_t stream) {
  (void)in_sizes; (void)n_in; (void)d_ws; (void)ws_size; (void)out_size;
  const float* x = (const float*)d_in[0];   // (16,256,224,224) f32
  float* out     = (float*)d_out;           // (16,256,7,7)     f32
  adaptive_pool7_kernel<<<dim3(NPLANE), dim3(224), 0, stream>>>(x, out);
}